// SphericalExpansion_36524401886020
// MI455X (gfx1250) — compile-verified
//
#include <hip/hip_runtime.h>

// ---------------------------------------------------------------------------
// Spherical expansion on MI455X (gfx1250).
//   out[c, m, q*8+n] = sum_{pairs p with center c} Y_m(u_p) * w_q(species) * R_n(r_p)
// Per-center 16x16xK GEMM via V_WMMA_F32_16X16X4_F32, K = pairs of that center.
// Pairs are binned by center directly into padded, 4-aligned segments so the
// WMMA inner loop is guard-free with affine addresses (no index indirection).
// The K-loop is 2x software-pipelined with independent accumulator chains.
// One wave per center -> atomic-free output stores.
// ---------------------------------------------------------------------------

namespace {
constexpr int   kSpecies = 4;
constexpr float kRcut    = 5.0f;
constexpr float kInvSig  = 1.6f;               // 1 / (R_CUT / N_MAX)
constexpr float kPi      = 3.14159265358979323846f;
}

typedef __attribute__((ext_vector_type(2))) float v2f;
typedef __attribute__((ext_vector_type(4))) float v4f;
typedef __attribute__((ext_vector_type(8))) float v8f;

__global__ void se_init_counts(int* counts, int* cursors, int n) {
  int i = blockIdx.x * blockDim.x + threadIdx.x;
  if (i < n) { counts[i] = 0; cursors[i] = 0; }
}

__global__ void se_count_pairs(const int* __restrict__ centers,
                               int* __restrict__ counts, int P) {
  int i = blockIdx.x * blockDim.x + threadIdx.x;
  if (i < P) atomicAdd(&counts[centers[i]], 1);
}

// Single-block exclusive scan over ceil4(count): each center gets a segment
// padded to a multiple of 4 pair-rows (so the WMMA K-loop needs no tail mask).
__global__ void se_exclusive_scan(const int* __restrict__ counts,
                                  int* __restrict__ offsets, int n) {
  __shared__ int sm[1024];
  __shared__ int carry;
  const int tid = threadIdx.x;
  if (tid == 0) carry = 0;
  __syncthreads();
  for (int base = 0; base < n; base += 1024) {
    int i = base + tid;
    int v = (i < n) ? ((counts[i] + 3) & ~3) : 0;   // padded count
    sm[tid] = v;
    __syncthreads();
    for (int off = 1; off < 1024; off <<= 1) {
      int t = (tid >= off) ? sm[tid - off] : 0;
      __syncthreads();
      sm[tid] += t;
      __syncthreads();
    }
    if (i < n) offsets[i] = carry + sm[tid] - v;    // exclusive prefix
    __syncthreads();
    if (tid == 0) carry += sm[1023];
    __syncthreads();
  }
}

// Zero the <=3 padding rows at the end of each center's segment.
__global__ void se_zero_padding(const int* __restrict__ counts,
                                const int* __restrict__ offsets,
                                float* __restrict__ Yws,
                                float* __restrict__ Fws, int nAtoms) {
  int i = blockIdx.x * blockDim.x + threadIdx.x;
  if (i >= nAtoms) return;
  int cnt  = counts[i];
  int cnt4 = (cnt + 3) & ~3;
  for (int k = offsets[i] + cnt; k < offsets[i] + cnt4; ++k) {
    v4f z = {};
    v4f* y = (v4f*)(Yws + (size_t)k * 16);
    v4f* f = (v4f*)(Fws + (size_t)k * 16);
    y[0] = z; y[1] = z; y[2] = z; y[3] = z;
    f[0] = z; f[1] = z; f[2] = z; f[3] = z;
  }
}

// Per-pair: Y[16] (real spherical harmonics, l<=3) and feat[16] = w_q * R_n,
// written DIRECTLY into the center-binned slot (binning fused here; no index
// array, so the WMMA pass reads at affine addresses).
__global__ void se_pair_features(const float* __restrict__ pos,
                                 const float* __restrict__ Wc,
                                 const int* __restrict__ centers,
                                 const int* __restrict__ neighbors,
                                 const int* __restrict__ species,
                                 const int* __restrict__ offsets,
                                 int* __restrict__ cursors,
                                 float* __restrict__ Yws,
                                 float* __restrict__ Fws, int P) {
  int i = blockIdx.x * blockDim.x + threadIdx.x;
  if (i >= P) return;
  int ci = centers[i], nj = neighbors[i];
  float dx = pos[3 * nj + 0] - pos[3 * ci + 0];
  float dy = pos[3 * nj + 1] - pos[3 * ci + 1];
  float dz = pos[3 * nj + 2] - pos[3 * ci + 2];
  float r  = sqrtf(dx * dx + dy * dy + dz * dz + 1e-12f);
  float inv = 1.0f / r;
  float x = dx * inv, y = dy * inv, z = dz * inv;
  float x2 = x * x, y2 = y * y, z2 = z * z;

  float fc = (r < kRcut) ? 0.5f * (cosf(kPi * r * (1.0f / kRcut)) + 1.0f) : 0.0f;

  int s = species[nj];
  float w0 = Wc[0 * kSpecies + s];
  float w1 = Wc[1 * kSpecies + s];

  float Y[16];
  Y[0]  = 0.28209479177387814f;
  Y[1]  = 0.4886025119029199f * y;
  Y[2]  = 0.4886025119029199f * z;
  Y[3]  = 0.4886025119029199f * x;
  Y[4]  = 1.0925484305920792f * x * y;
  Y[5]  = 1.0925484305920792f * y * z;
  Y[6]  = 0.31539156525252005f * (3.0f * z2 - 1.0f);
  Y[7]  = 1.0925484305920792f * x * z;
  Y[8]  = 0.5462742152960396f * (x2 - y2);
  Y[9]  = 0.5900435899266435f * y * (3.0f * x2 - y2);
  Y[10] = 2.890611442640554f * x * y * z;
  Y[11] = 0.4570457994644658f * y * (5.0f * z2 - 1.0f);
  Y[12] = 0.3731763325901154f * z * (5.0f * z2 - 3.0f);
  Y[13] = 0.4570457994644658f * x * (5.0f * z2 - 1.0f);
  Y[14] = 1.445305721320277f * z * (x2 - y2);
  Y[15] = 0.5900435899266435f * x * (x2 - 3.0f * y2);

  float rad[8];
#pragma unroll
  for (int n = 0; n < 8; ++n) {
    float cn = kRcut * (float)n * (1.0f / 7.0f);
    float t  = (r - cn) * kInvSig;
    rad[n] = __expf(-0.5f * t * t) * fc;
  }

  // Binned slot for this pair.
  int slot = offsets[ci] + atomicAdd(&cursors[ci], 1);

  float* Yo = Yws + (size_t)slot * 16;
  float* Fo = Fws + (size_t)slot * 16;
#pragma unroll
  for (int m = 0; m < 16; ++m) Yo[m] = Y[m];
#pragma unroll
  for (int n = 0; n < 8; ++n) { Fo[n] = w0 * rad[n]; Fo[8 + n] = w1 * rad[n]; }
}

// One wave per center: C[16x16] += A(16x4:Y) x B(4x16:feat), 4 pairs/WMMA.
// Lane mapping per ISA 32-bit A/B layouts: lane = 16*half + row,
//   A: vgpr0 = Y[row] of pair (2*half), vgpr1 = Y[row] of pair (2*half+1)
//   B: vgpr0 = F[row] of pair (2*half), vgpr1 = F[row] of pair (2*half+1)
// Segments are padded to 4 with zero rows -> no guards in the K-loop.
// 2x unrolled with two independent accumulator chains to overlap loads/WMMAs.
__global__ void se_wmma_accumulate(const float* __restrict__ Yws,
                                   const float* __restrict__ Fws,
                                   const int* __restrict__ offsets,
                                   const int* __restrict__ counts,
                                   float* __restrict__ out, int nAtoms) {
  int wave = (blockIdx.x * blockDim.x + threadIdx.x) >> 5;   // wave32
  int lane = threadIdx.x & 31;
  if (wave >= nAtoms) return;                                // wave-uniform exit

  // Wave-uniform by construction; force into SGPRs for scalar loop control.
  const int begin = __builtin_amdgcn_readfirstlane(offsets[wave]);
  const int cnt4  = __builtin_amdgcn_readfirstlane((counts[wave] + 3) & ~3);
  const int half  = lane >> 4;
  const int row   = lane & 15;

  // Lane-local affine bases: pair-row k lives at base[k*16].
  const float* Yb = Yws + ((size_t)begin + (size_t)(2 * half)) * 16 + row;
  const float* Fb = Fws + ((size_t)begin + (size_t)(2 * half)) * 16 + row;

  v8f c0 = {};
  v8f c1 = {};
  int k = 0;
  // Main loop: 8 pairs / iteration, 8 loads clause-issued, 2 independent WMMAs.
  for (; k + 8 <= cnt4; k += 8) {
    const int i0 = k * 16;
    const int i1 = (k + 4) * 16;
    v2f a0, b0, a1, b1;
    a0.x = Yb[i0];  a0.y = Yb[i0 + 16];
    b0.x = Fb[i0];  b0.y = Fb[i0 + 16];
    a1.x = Yb[i1];  a1.y = Yb[i1 + 16];
    b1.x = Fb[i1];  b1.y = Fb[i1 + 16];
    c0 = __builtin_amdgcn_wmma_f32_16x16x4_f32(
        false, a0, false, b0, (short)0, c0, false, false);
    c1 = __builtin_amdgcn_wmma_f32_16x16x4_f32(
        false, a1, false, b1, (short)0, c1, false, false);
  }
  // 4-aligned tail (at most one chunk).
  if (k < cnt4) {
    const int i0 = k * 16;
    v2f a0, b0;
    a0.x = Yb[i0];  a0.y = Yb[i0 + 16];
    b0.x = Fb[i0];  b0.y = Fb[i0 + 16];
    c0 = __builtin_amdgcn_wmma_f32_16x16x4_f32(
        false, a0, false, b0, (short)0, c0, false, false);
  }
  v8f c = c0 + c1;   // each pair contributed to exactly one chain

  // C layout: vgpr r -> M = r + 8*half, N = row. out[atom][m][feature].
  float* o = out + (size_t)wave * 256;
#pragma unroll
  for (int r = 0; r < 8; ++r) {
    int M = r + 8 * half;
    o[M * 16 + row] = c[r];
  }
}

extern "C" void kernel_launch(void* const* d_in, const int* in_sizes, int n_in,
                              void* d_out, int out_size, void* d_ws, size_t ws_size,
                              hipStream_t stream) {
  const float* pos       = (const float*)d_in[0];   // [A,3]
  const float* Wc        = (const float*)d_in[1];   // [2,4]
  const int*   centers   = (const int*)d_in[2];     // [P]
  const int*   neighbors = (const int*)d_in[3];     // [P]
  const int*   species   = (const int*)d_in[4];     // [A]
  const int P = in_sizes[2];
  const int A = in_sizes[4];
  const size_t Pcap = (size_t)P + 3u * (size_t)A;   // padded pair-row capacity

  // Carve workspace (aligned to 256B). Total ~66 MB.
  char* w = (char*)d_ws;
  auto carve = [&](size_t bytes) {
    char* p = w;
    w += (bytes + 255) & ~(size_t)255;
    return p;
  };
  int*   counts  = (int*)carve((size_t)A * sizeof(int));
  int*   cursors = (int*)carve((size_t)A * sizeof(int));
  int*   offsets = (int*)carve((size_t)A * sizeof(int));
  float* Yws     = (float*)carve(Pcap * 16 * sizeof(float));
  float* Fws     = (float*)carve(Pcap * 16 * sizeof(float));

  const int TB = 256;
  se_init_counts<<<(A + TB - 1) / TB, TB, 0, stream>>>(counts, cursors, A);
  se_count_pairs<<<(P + TB - 1) / TB, TB, 0, stream>>>(centers, counts, P);
  se_exclusive_scan<<<1, 1024, 0, stream>>>(counts, offsets, A);
  se_zero_padding<<<(A + TB - 1) / TB, TB, 0, stream>>>(counts, offsets, Yws, Fws, A);
  se_pair_features<<<(P + TB - 1) / TB, TB, 0, stream>>>(pos, Wc, centers, neighbors,
                                                         species, offsets, cursors,
                                                         Yws, Fws, P);
  // 8 waves per 256-thread block, one wave per center atom.
  se_wmma_accumulate<<<(A + 7) / 8, TB, 0, stream>>>(Yws, Fws, offsets, counts,
                                                     (float*)d_out, A);
}